// FLUX_47244640256618
// MI455X (gfx1250) — compile-verified
//
#include <hip/hip_runtime.h>
#include <hip/hip_bf16.h>
#include <math.h>

// MI455X (gfx1250) — bandwidth-bound grouped tiny-MLP + GEMM fusion.
// 550MB x-stream @23.3TB/s => ~24us floor; f32 WMMA (2048 FLOP/instr) keeps
// the 2.6 GFLOP of compute far below that floor. Hot-loop loads are branchless
// (cndmask, not EXEC-masked regions) so A-loads form one 13-deep clause and
// B-loads pair into ds_load_2addr_b64 pipelined between WMMAs. The hidden-dim
// reduction is done with DPP16 VALU adds (no ds_bpermute / DS-pipe pressure).

typedef float v2f __attribute__((ext_vector_type(2)));
typedef float v8f __attribute__((ext_vector_type(8)));

namespace {
constexpr int kB      = 16384;
constexpr int kM      = 168;     // modules
constexpr int kG      = 50;      // genes
constexpr int kGP     = 52;      // K padded to WMMA multiple of 4
constexpr int kH      = 8;       // hidden
constexpr int kNC     = 70;      // comps
constexpr int kChunk  = 42;      // modules staged in LDS at a time (168/4)
constexpr int kNChunk = 4;
constexpr int kCmStr  = 72;      // padded cmT row stride
constexpr int kRowLen = kM * kG; // 8400 floats per batch row
}

__device__ __forceinline__ float tanh_fast(float v) {
#if __has_builtin(__builtin_amdgcn_tanhf)
  return __builtin_amdgcn_tanhf(v);   // native v_tanh_f32 (TRANS op)
#else
  return tanhf(v);
#endif
}

// v += lane-permuted(v) via DPP16; CTRL is an immediate dpp_ctrl encoding.
template <int CTRL>
__device__ __forceinline__ float dpp_add(float v) {
  int p = __builtin_amdgcn_update_dpp(0, __float_as_int(v), CTRL,
                                      /*row_mask=*/0xf, /*bank_mask=*/0xf,
                                      /*bound_ctrl=*/true);
  return v + __int_as_float(p);
}

// Sum lanes {0..7} (lanes 8..15 hold zeros) and broadcast to all 16 lanes of
// each half-wave row. Equivalent to shfl_xor 1,2,4,8 for this data pattern.
__device__ __forceinline__ float reduce8_bcast16(float v) {
  v = dpp_add<0xB1>(v);   // quad_perm(1,0,3,2)  : xor 1
  v = dpp_add<0x4E>(v);   // quad_perm(2,3,0,1)  : xor 2
  v = dpp_add<0x141>(v);  // ROW_HALF_MIRROR     : cross-quad within 8
  v = dpp_add<0x140>(v);  // ROW_MIRROR          : cross-8 within 16
  return v;
}

__global__ __launch_bounds__(256)
void FLUX_47244640256618_kernel(const float* __restrict__ x,
                                const float* __restrict__ cmMat,
                                const float* __restrict__ scale,
                                const int*   __restrict__ cell_id,
                                const float* __restrict__ W1,
                                const float* __restrict__ W2,
                                const float* __restrict__ b2,
                                float* __restrict__ mout,
                                float* __restrict__ cout)
{
  __shared__ __align__(16) float w1buf[kChunk * kH * kGP];  // 17472 f = 69.9KB
  __shared__ __align__(16) float zrow[kGP];                 // zero row for n>=8
  __shared__ float w2buf[kM * 16];                          // padded: 10.8KB
  __shared__ float b2buf[kM];
  __shared__ float cmT[kM * kCmStr + 16];                   // +16 pad: 48.4KB
  // total ~130KB -> 2 workgroups per WGP (320KB LDS), 4 waves/SIMD

  const int tid  = threadIdx.x;
  const int wave = tid >> 5;
  const int lane = tid & 31;
  const int half = lane >> 4;      // 0: K+0/K+1 & rows 0-7 ; 1: K+2/K+3 & rows 8-15
  const int n    = lane & 15;      // A: row ; B/C/D: column
  const int b0   = (blockIdx.x * 8 + wave) * 16;

  // ---- one-time staging (covered by first chunk barrier) ----
  for (int i = tid; i < kGP; i += 256) zrow[i] = 0.f;
  for (int i = tid; i < kChunk * kH; i += 256) {            // K-pad slots, once
    float2 z; z.x = 0.f; z.y = 0.f;
    *(float2*)(w1buf + i * kGP + kG) = z;
  }
  for (int i = tid; i < kM * 16; i += 256) {                // W2 padded to 16
    int mod = i >> 4, h = i & 15;
    w2buf[i] = (h < kH) ? W2[mod * kH + h] : 0.f;
  }
  for (int i = tid; i < kM; i += 256) b2buf[i] = b2[i];
  for (int i = tid; i < kM * kNC; i += 256) {
    int mod = i / kNC, col = i - mod * kNC;
    cmT[mod * kCmStr + col] = cmMat[col * kM + mod];        // cmT[mod][comp]
  }

  // per-lane scale for the 8 output rows this lane touches (row = b0+j+8*half)
  float msc[8];
#pragma unroll
  for (int j = 0; j < 8; ++j)
    msc[j] = scale[cell_id[b0 + j + 8 * half]];

  float cacc[5][8];
#pragma unroll
  for (int t = 0; t < 5; ++t)
#pragma unroll
    for (int j = 0; j < 8; ++j) cacc[t][j] = 0.f;
  float mpack[8];
#pragma unroll
  for (int j = 0; j < 8; ++j) mpack[j] = 0.f;

  const float* xrow = x + (size_t)(b0 + n) * kRowLen;  // A-matrix row, this lane

  for (int chunk = 0; chunk < kNChunk; ++chunk) {
    __syncthreads();  // previous chunk consumers done
    {
      // W1 chunk: 16800 floats, copied as coalesced b64 with K re-pad 50->52
      const float* src = W1 + chunk * (kChunk * kH * kG);
      for (int i = tid; i < (kChunk * kH * kG) / 2; i += 256) {
        const int s = 2 * i;
        const int row = s / kG;            // 0..335
        const int col = s - row * kG;      // even, pairs never straddle a row
        *(float2*)(w1buf + row * kGP + col) = *(const float2*)(src + s);
      }
    }
    __syncthreads();  // chunk staged

    for (int lm = 0; lm < kChunk; ++lm) {
      const int gmod = chunk * kChunk + lm;

      // pull next module's x segment into cache (global_prefetch_b8)
      {
        const int pm = (gmod + 1 < kM) ? (gmod + 1) : gmod;
        __builtin_prefetch(xrow + pm * kG, 0, 3);
      }

      // ---- B regs: W1[gmod]^T as 52x16; lanes n>=8 read the zero row ----
      const float* wb = (n < kH) ? (w1buf + lm * (kH * kGP) + n * kGP) : zrow;
      float2 bv[13];
#pragma unroll
      for (int kk = 0; kk < 13; ++kk)
        bv[kk] = *(const float2*)(wb + 4 * kk + 2 * half);   // ds_load_2addr_b64

      // ---- A regs + 13 chained f32 WMMAs: D = x_tile(16x52) * B(52x16) ----
      const float* xa = xrow + gmod * kG;
      v8f acc = {0.f, 0.f, 0.f, 0.f, 0.f, 0.f, 0.f, 0.f};
#pragma unroll
      for (int kk = 0; kk < 13; ++kk) {
        const int k0 = 4 * kk + 2 * half;
        const bool kvalid = (k0 < kG);                        // false only kk=12,half=1
        const float2 a = *(const float2*)(xa + (kvalid ? k0 : kG - 4));
        v2f av;
        av.x = kvalid ? a.x : 0.f;
        av.y = kvalid ? a.y : 0.f;
        v2f bb = {bv[kk].x, bv[kk].y};
        acc = __builtin_amdgcn_wmma_f32_16x16x4_f32(
            /*neg_a=*/false, av, /*neg_b=*/false, bb,
            /*c_mod=*/(short)0, acc, /*reuse_a=*/false, /*reuse_b=*/false);
      }

      // ---- tanhshrink(h) * W2, DPP reduce over the 8 hidden cols ----
      const float w2v = w2buf[gmod * 16 + n];                 // 0 for n>=8
      float p[8];
#pragma unroll
      for (int j = 0; j < 8; ++j) {
        float hv = acc[j];
        hv = hv - tanh_fast(hv);       // tanhshrink; cols>=8 are exactly 0
        p[j] = hv * w2v;
      }
#pragma unroll
      for (int j = 0; j < 8; ++j)
        p[j] = reduce8_bcast16(p[j]);  // row-sum broadcast to all 16 lanes/half

      const float bb2 = b2buf[gmod];
      float mv[8];
#pragma unroll
      for (int j = 0; j < 8; ++j) {
        float t = p[j] + bb2;
        t = t - tanh_fast(t);          // tanhshrink
        mv[j] = t * msc[j];            // per-cell scale
      }

      // ---- pack m into lanes by (gmod & 15); coalesced store per 16 modules ----
      const bool keep = (n == (gmod & 15));
#pragma unroll
      for (int j = 0; j < 8; ++j) mpack[j] = keep ? mv[j] : mpack[j];
      if ((gmod & 15) == 15 || gmod == kM - 1) {              // uniform branch
        const int gbase = gmod & ~15;
        const int ncols = gmod - gbase + 1;   // 16, or 8 for last group
        if (n < ncols) {
#pragma unroll
          for (int j = 0; j < 8; ++j)
            mout[(size_t)(b0 + j + 8 * half) * kM + gbase + n] = mpack[j];
        }
      }

      // ---- accumulate c[row][comp] += m[row][gmod] * cmMat[comp][gmod] ----
      // cols 70..79 accumulate dead values (never stored); row is padded so
      // the read stays in-bounds.
#pragma unroll
      for (int t = 0; t < 5; ++t) {
        const float cmv = cmT[gmod * kCmStr + t * 16 + n];
#pragma unroll
        for (int j = 0; j < 8; ++j) cacc[t][j] = fmaf(mv[j], cmv, cacc[t][j]);
      }
    }
  }

  // ---- write c tile (16 x 70) ----
#pragma unroll
  for (int t = 0; t < 5; ++t) {
    const int col = t * 16 + n;
    if (col < kNC) {
#pragma unroll
      for (int j = 0; j < 8; ++j)
        cout[(size_t)(b0 + j + 8 * half) * kNC + col] = cacc[t][j];
    }
  }
}

extern "C" void kernel_launch(void* const* d_in, const int* in_sizes, int n_in,
                              void* d_out, int out_size, void* d_ws, size_t ws_size,
                              hipStream_t stream) {
  // setup_inputs order: x, n_modules, n_genes, n_comps, cmMat, scale, cell_id, W1, W2, b2
  const float* x       = (const float*)d_in[0];
  const float* cmMat   = (const float*)d_in[4];
  const float* scalep  = (const float*)d_in[5];
  const int*   cell_id = (const int*)d_in[6];
  const float* W1      = (const float*)d_in[7];
  const float* W2      = (const float*)d_in[8];
  const float* b2      = (const float*)d_in[9];

  float* mout = (float*)d_out;                     // [16384, 168]
  float* cout = mout + (size_t)kB * kM;            // [16384, 70]

  // 128 blocks x 8 waves x 16-row tiles = 16384 rows
  FLUX_47244640256618_kernel<<<dim3(128), dim3(256), 0, stream>>>(
      x, cmMat, scalep, cell_id, W1, W2, b2, mout, cout);
}